// LLRDenoiser_23759759081974
// MI455X (gfx1250) — compile-verified
//
#include <hip/hip_runtime.h>
#include <math.h>
#include <stdint.h>

// ---------------------------------------------------------------------------
// LLR denoiser for (B,C,H,W)=(2,16,512,512), P=8, STRIDE=4, THS=0.1
// One wave32 = one patch. SVD via Gram matrix + Jacobi eigensolve.
//  - patch gather: TDM tensor_load_to_lds (3D tile, LDS pad -> stride 68)
//  - GEMM stages:  V_WMMA_F32_16X16X4_F32 (full fp32 WMMA on CDNA5)
// ---------------------------------------------------------------------------

typedef float v2f __attribute__((ext_vector_type(2)));
typedef float v8f __attribute__((ext_vector_type(8)));
typedef unsigned int u32x4 __attribute__((ext_vector_type(4)));
typedef int i32x4 __attribute__((ext_vector_type(4)));
typedef int i32x8 __attribute__((ext_vector_type(8)));

constexpr int Bc = 2, Cc = 16, Hc = 512, Wc = 512;
constexpr int Pc = 8, Sc = 4;
constexpr int NH = (Hc - Pc) / Sc + 1;     // 127
constexpr int NW = (Wc - Pc) / Sc + 1;     // 127
constexpr int NPATCH = Bc * NH * NW;       // 32258
constexpr float THS = 0.1f;

constexpr int WAVES = 8;                   // waves per block
constexpr int SM = 68;                     // LDS stride for 16x64 M/T tile
                                           // (TDM pad: 4 DW after every 64 DW)
constexpr int SG = 17;                     // LDS stride for 16x16 G/U
constexpr int FLW = 16 * SM + 16 * SG + 16 * SG + 32; // floats per wave = 1664

__device__ __forceinline__ void wave_lds_sync() {
  // wave-private LDS data: lanes are lockstep, just drain DS counter and
  // stop compiler/scheduler motion across this point.
  asm volatile("s_wait_dscnt 0x0" ::: "memory");
  __builtin_amdgcn_wave_barrier();
}

__device__ __forceinline__ v8f wmma4(v2f a, v2f b, v8f c) {
  // D = A(16x4,f32) * B(4x16,f32) + C(16x16,f32)
  return __builtin_amdgcn_wmma_f32_16x16x4_f32(
      /*neg_a=*/false, a, /*neg_b=*/false, b,
      /*c_mod=*/(short)0, c, /*reuse_a=*/false, /*reuse_b=*/false);
}

__device__ __forceinline__ int rr_player(int rnd, int pos) {
  // round-robin tournament schedule for 16 players, 15 rounds
  return (pos == 0) ? 0 : ((rnd + pos - 1) % 15) + 1;
}

__global__ __launch_bounds__(WAVES * 32)
void llr_patch_kernel(const float* __restrict__ x, float* __restrict__ out) {
  __shared__ float smem[WAVES * FLW];

  const int lane = threadIdx.x & 31;
  const int wave = threadIdx.x >> 5;
  const int patch = blockIdx.x * WAVES + wave;
  if (patch >= NPATCH) return;                 // uniform per wave

  const int b   = patch / (NH * NW);
  const int rem = patch % (NH * NW);
  const int h0  = (rem / NW) * Sc;
  const int w0  = (rem % NW) * Sc;

  float* Ml  = &smem[wave * FLW];              // 16 x SM : patch matrix M, later T
  float* Gl  = Ml + 16 * SM;                   // 16 x SG : Gram G, later Us
  float* Ul  = Gl + 16 * SG;                   // 16 x SG : eigenvectors U
  float* csA = Ul + 16 * SG;                   // 8 cos
  float* snA = csA + 8;                        // 8 sin
  float* rat = snA + 8;                        // 16 ratios

  const size_t plane = (size_t)Hc * Wc;
  const float* xb = x + (size_t)b * Cc * plane;

  // ---- 1) TDM: DMA the 8x8x16 patch tile into LDS as M[ch][8*dy+dx] -------
  // D# group0: count=1 | lds_addr | global_addr(57b) | type=2
  {
    const float* gsrc = xb + (size_t)h0 * Wc + w0;
    const uint64_t ga = (uint64_t)(uintptr_t)gsrc;
    const uint32_t ga_lo = __builtin_amdgcn_readfirstlane((uint32_t)ga);
    const uint32_t ga_hi = __builtin_amdgcn_readfirstlane((uint32_t)(ga >> 32));
    const uint32_t lds_a =
        __builtin_amdgcn_readfirstlane((uint32_t)(uintptr_t)Ml);

    u32x4 g0;
    g0[0] = 1u;                                       // count=1, user mode
    g0[1] = lds_a;                                    // LDS byte address
    g0[2] = ga_lo;                                    // global_addr[31:0]
    g0[3] = (ga_hi & 0x01FFFFFFu) | 0x80000000u;      // addr[56:32] | type=2

    // D# group1: data_size=4B(2), pad_enable, pad_interval=64DW(5),
    // pad_amount=4DW(3); tensor_dim0=512, tensor_dim1=512,
    // tile=(8,8,16); dim0_stride=512, dim1_stride=512*512
    i32x8 g1;
    g1[0] = (int)((2u << 16) | (1u << 20) | (5u << 22) | (3u << 25));
    g1[1] = (int)((uint32_t)(Wc & 0xFFFF) << 16);     // tensor_dim0 low16
    g1[2] = (int)(((uint32_t)Wc >> 16) | ((uint32_t)(Hc & 0xFFFF) << 16));
    g1[3] = (int)(((uint32_t)Hc >> 16) | ((uint32_t)Pc << 16)); // tile_dim0=8
    g1[4] = (int)((uint32_t)Pc | ((uint32_t)Cc << 16));         // tile 8 x 16
    g1[5] = (int)Wc;                                  // dim0_stride[31:0]
    g1[6] = (int)(((uint32_t)plane & 0xFFFFu) << 16); // dim1_stride[15:0]
    g1[7] = (int)((uint32_t)(plane >> 16));           // dim1_stride[47:16]

    i32x4 g2;                                          // 3D tile: dim2=C
    g2[0] = Cc; g2[1] = 0; g2[2] = 0; g2[3] = 0;
    i32x4 g3;
    g3[0] = 0; g3[1] = 0; g3[2] = 0; g3[3] = 0;
    i32x8 g4;                                          // extra group (clang-23
    g4[0] = 0; g4[1] = 0; g4[2] = 0; g4[3] = 0;        //  6-arg form), unused
    g4[4] = 0; g4[5] = 0; g4[6] = 0; g4[7] = 0;

    __builtin_amdgcn_tensor_load_to_lds(g0, g1, g2, g3, g4, /*cpol=*/0);
  }

  // U = I (overlaps with the TDM transfer)
  #pragma unroll
  for (int i = 0; i < 8; ++i) {
    const int e = lane + 32 * i;               // 0..255
    const int r = e >> 4, c = e & 15;
    Ul[r * SG + c] = (r == c) ? 1.0f : 0.0f;
  }
  __builtin_amdgcn_s_wait_tensorcnt((short)0);  // M tile landed in LDS
  wave_lds_sync();

  const int n16    = lane & 15;                // row/col owned by lane
  const int khalf  = (lane >> 4) << 1;         // 0 or 2 (A/B frag K select)
  const int rowoff = (lane >> 4) << 3;         // 0 or 8 (C/D frag M select)

  // ---- 2) Gram G = M * M^T (K=64 -> 16 chained WMMA f32 16x16x4) ----------
  v8f g = {0.f, 0.f, 0.f, 0.f, 0.f, 0.f, 0.f, 0.f};
  #pragma unroll
  for (int kk = 0; kk < 16; ++kk) {
    const int kb = kk * 4 + khalf;
    v2f a;
    a.x = Ml[n16 * SM + kb];
    a.y = Ml[n16 * SM + kb + 1];
    // B[k][n] = M[n][k] == same per-lane values as A -> reuse
    g = wmma4(a, a, g);
  }
  #pragma unroll
  for (int r = 0; r < 8; ++r) Gl[(r + rowoff) * SG + n16] = g[r];
  wave_lds_sync();

  // ---- 3) Jacobi eigensolve of G (16x16 sym), accumulate U ----------------
  for (int sweep = 0; sweep < 8; ++sweep) {
    for (int rnd = 0; rnd < 15; ++rnd) {
      if (lane < 8) {                          // one rotation per lane
        int p = rr_player(rnd, lane);
        int q = rr_player(rnd, 15 - lane);
        if (p > q) { int t = p; p = q; q = t; }
        const float app = Gl[p * SG + p];
        const float aqq = Gl[q * SG + q];
        const float apq = Gl[p * SG + q];
        float c0 = 1.0f, s0 = 0.0f;
        if (fabsf(apq) > 1e-12f) {
          const float tau = (aqq - app) / (2.0f * apq);
          float t = 1.0f / (fabsf(tau) + sqrtf(1.0f + tau * tau));
          t = (tau >= 0.0f) ? t : -t;
          c0 = 1.0f / sqrtf(1.0f + t * t);
          s0 = t * c0;
        }
        csA[lane] = c0;
        snA[lane] = s0;
      }
      wave_lds_sync();
      // column pass: G <- G*J and U <- U*J  (8 disjoint pairs, 256 tasks)
      #pragma unroll
      for (int i = 0; i < 8; ++i) {
        const int task = lane + 32 * i;        // 0..255
        const int j    = task & 7;
        const int row  = (task >> 3) & 15;
        float* Mp = (task >= 128) ? Ul : Gl;
        int p = rr_player(rnd, j);
        int q = rr_player(rnd, 15 - j);
        if (p > q) { int t = p; p = q; q = t; }
        const float c0 = csA[j], s0 = snA[j];
        const float vp = Mp[row * SG + p];
        const float vq = Mp[row * SG + q];
        Mp[row * SG + p] = c0 * vp - s0 * vq;
        Mp[row * SG + q] = s0 * vp + c0 * vq;
      }
      wave_lds_sync();
      // row pass: G <- J^T * G  (128 tasks)
      #pragma unroll
      for (int i = 0; i < 4; ++i) {
        const int task = lane + 32 * i;        // 0..127
        const int j    = task & 7;
        const int col  = (task >> 3) & 15;
        int p = rr_player(rnd, j);
        int q = rr_player(rnd, 15 - j);
        if (p > q) { int t = p; p = q; q = t; }
        const float c0 = csA[j], s0 = snA[j];
        const float vp = Gl[p * SG + col];
        const float vq = Gl[q * SG + col];
        Gl[p * SG + col] = c0 * vp - s0 * vq;
        Gl[q * SG + col] = s0 * vp + c0 * vq;
      }
      wave_lds_sync();
    }
  }

  // ---- 4) singular-value soft-threshold ratios; Us = U*diag(ratio) -> Gl --
  if (lane < 16) {
    const float lam = Gl[lane * SG + lane];
    const float s   = sqrtf(lam > 0.0f ? lam : 0.0f);
    float st = s - THS;
    st = (st > 0.0f) ? st : 0.0f;
    rat[lane] = (s > 1e-12f) ? (st / s) : 0.0f;
  }
  wave_lds_sync();
  #pragma unroll
  for (int i = 0; i < 8; ++i) {
    const int e = lane + 32 * i;
    const int r = e >> 4, c = e & 15;
    Gl[r * SG + c] = Ul[r * SG + c] * rat[c];  // Us overwrites G
  }
  wave_lds_sync();

  // ---- 5) T = U^T * M  (overwrite M tile-by-tile, 4 tiles x K=16) ---------
  #pragma unroll
  for (int t = 0; t < 4; ++t) {
    v8f acc = {0.f, 0.f, 0.f, 0.f, 0.f, 0.f, 0.f, 0.f};
    #pragma unroll
    for (int kk = 0; kk < 4; ++kk) {
      const int kb = kk * 4 + khalf;
      v2f a;                                   // A[m][k] = U[k][m]
      a.x = Ul[kb * SG + n16];
      a.y = Ul[(kb + 1) * SG + n16];
      v2f bf;                                  // B[k][n] = M[k][16t+n]
      bf.x = Ml[kb * SM + t * 16 + n16];
      bf.y = Ml[(kb + 1) * SM + t * 16 + n16];
      acc = wmma4(a, bf, acc);
    }
    wave_lds_sync();
    #pragma unroll
    for (int r = 0; r < 8; ++r) Ml[(r + rowoff) * SM + t * 16 + n16] = acc[r];
    wave_lds_sync();
  }

  // ---- 6) M' = Us * T, scatter-add into output ----------------------------
  float* outb = out + (size_t)b * Cc * plane;
  #pragma unroll
  for (int t = 0; t < 4; ++t) {
    v8f acc = {0.f, 0.f, 0.f, 0.f, 0.f, 0.f, 0.f, 0.f};
    #pragma unroll
    for (int kk = 0; kk < 4; ++kk) {
      const int kb = kk * 4 + khalf;
      v2f a;                                   // A[m][k] = Us[m][k]
      a.x = Gl[n16 * SG + kb];
      a.y = Gl[n16 * SG + kb + 1];
      v2f bf;                                  // B[k][n] = T[k][16t+n]
      bf.x = Ml[kb * SM + t * 16 + n16];
      bf.y = Ml[(kb + 1) * SM + t * 16 + n16];
      acc = wmma4(a, bf, acc);
    }
    const int col = t * 16 + n16;              // patch pixel index 0..63
    const int hh  = h0 + (col >> 3);
    const int ww  = w0 + (col & 7);
    #pragma unroll
    for (int r = 0; r < 8; ++r) {
      const int ch = r + rowoff;
      atomicAdd(&outb[(size_t)ch * plane + (size_t)hh * Wc + ww], acc[r]);
    }
  }
}

// coverage count along one axis (stride 4, patch 8, last start = 504)
__device__ __forceinline__ int coverage(int n) {
  int lo = n - (Pc - 1); if (lo < 0) lo = 0;
  int hi = n; if (hi > Hc - Pc) hi = Hc - Pc;
  return hi / Sc - (lo + Sc - 1) / Sc + 1;
}

__global__ __launch_bounds__(256)
void llr_norm_kernel(float* __restrict__ out) {
  const size_t total = (size_t)Bc * Cc * Hc * Wc;
  const size_t idx = (size_t)blockIdx.x * blockDim.x + threadIdx.x;
  if (idx >= total) return;
  const int w = (int)(idx % Wc);
  const int h = (int)((idx / Wc) % Hc);
  out[idx] *= 1.0f / (float)(coverage(h) * coverage(w));
}

extern "C" void kernel_launch(void* const* d_in, const int* in_sizes, int n_in,
                              void* d_out, int out_size, void* d_ws, size_t ws_size,
                              hipStream_t stream) {
  (void)in_sizes; (void)n_in; (void)d_ws; (void)ws_size;
  const float* x = (const float*)d_in[0];
  float* out = (float*)d_out;

  (void)hipMemsetAsync(out, 0, (size_t)out_size * sizeof(float), stream);

  const int blocks = (NPATCH + WAVES - 1) / WAVES;           // 4033
  llr_patch_kernel<<<blocks, WAVES * 32, 0, stream>>>(x, out);

  const size_t total = (size_t)Bc * Cc * Hc * Wc;            // 8388608
  const int nblocks = (int)((total + 255) / 256);
  llr_norm_kernel<<<nblocks, 256, 0, stream>>>(out);
}